// RNN_57423712747897
// MI455X (gfx1250) — compile-verified
//
#include <hip/hip_runtime.h>
#include <math.h>

typedef __attribute__((ext_vector_type(2))) float v2f;
typedef __attribute__((ext_vector_type(8))) float v8f;

#define B_SZ   256
#define T_SZ   1024
#define F_TR   128
#define F_ST   32
#define H_SZ   3
#define OUT_SZ 128
#define KFC    (F_ST + H_SZ)   // 35, padded to 36 in the FC kernel

// ---------------------------------------------------------------------------
// Kernel A: x_proj[t][b][h] = tr_input[b][t][:] . W_ih[h][:] + b_ih[h] + b_hh[h]
// GEMM [B*T,128] x [128,3] via V_WMMA_F32_16X16X4_F32, one wave per 16-row tile.
// ---------------------------------------------------------------------------
__global__ __launch_bounds__(256) void xproj_wmma_kernel(
    const float* __restrict__ tr,    // [B, T, F_TR]
    const float* __restrict__ W_ih,  // [H, F_TR]
    const float* __restrict__ b_ih,  // [H]
    const float* __restrict__ b_hh,  // [H]
    float* __restrict__ xproj)       // [T, B, H]
{
    const int lane = threadIdx.x & 31;
    const int wave = threadIdx.x >> 5;
    const int tile = blockIdx.x * 8 + wave;        // M-tile over B*T rows
    const int rowBase = tile * 16;

    const int half = lane >> 4;                    // K-half select
    const int l16  = lane & 15;                    // A: M row / B: N col

    const float* arow = tr + (size_t)(rowBase + l16) * F_TR;
    // branchless B operand: clamp row index, mask value (keeps EXEC all-1s)
    const int   nclamp = (l16 < H_SZ) ? l16 : 0;
    const float nmask  = (l16 < H_SZ) ? 1.0f : 0.0f;
    const float* wrow  = W_ih + nclamp * F_TR;

    v8f c = {};
    #pragma unroll
    for (int k0 = 0; k0 < F_TR; k0 += 4) {
        const int k = k0 + 2 * half;
        v2f a, b;
        a.x = arow[k];
        a.y = arow[k + 1];
        b.x = wrow[k]     * nmask;
        b.y = wrow[k + 1] * nmask;
        c = __builtin_amdgcn_wmma_f32_16x16x4_f32(
            /*neg_a=*/false, a, /*neg_b=*/false, b,
            /*c_mod=*/(short)0, c, /*reuse_a=*/false, /*reuse_b=*/false);
    }

    // D layout: VGPR i -> M = i + 8*half, N = l16.  Only N < 3 is real output.
    if (l16 < H_SZ) {
        const float bias = b_ih[l16] + b_hh[l16];
        #pragma unroll
        for (int i = 0; i < 8; ++i) {
            const int r  = rowBase + i + 8 * half;   // global row = b*T + t
            const int bb = r >> 10;                  // r / T
            const int tt = r & (T_SZ - 1);           // r % T
            xproj[((size_t)tt * B_SZ + bb) * H_SZ + l16] = c[i] + bias;
        }
    }
}

// ---------------------------------------------------------------------------
// Kernel B: serial tanh-RNN scan over T, one lane per batch element.
// h_new = tanh(xp + h @ W_hh^T); accumulate time-mean of h.
// ---------------------------------------------------------------------------
__global__ __launch_bounds__(256) void rnn_scan_kernel(
    const float* __restrict__ xproj,  // [T, B, H]
    const float* __restrict__ W_hh,   // [H, H]
    float* __restrict__ hmean)        // [B, H]
{
    const int b = threadIdx.x;        // grid = 1 block of 256

    const float w00 = W_hh[0], w01 = W_hh[1], w02 = W_hh[2];
    const float w10 = W_hh[3], w11 = W_hh[4], w12 = W_hh[5];
    const float w20 = W_hh[6], w21 = W_hh[7], w22 = W_hh[8];

    float h0 = 0.f, h1 = 0.f, h2 = 0.f;
    float s0 = 0.f, s1 = 0.f, s2 = 0.f;

    const float* xp = xproj + (size_t)b * H_SZ;
    #pragma unroll 2
    for (int t = 0; t < T_SZ; ++t) {
        const float x0 = xp[0], x1 = xp[1], x2 = xp[2];
        xp += B_SZ * H_SZ;
        const float n0 = tanhf(fmaf(w02, h2, fmaf(w01, h1, fmaf(w00, h0, x0))));
        const float n1 = tanhf(fmaf(w12, h2, fmaf(w11, h1, fmaf(w10, h0, x1))));
        const float n2 = tanhf(fmaf(w22, h2, fmaf(w21, h1, fmaf(w20, h0, x2))));
        h0 = n0; h1 = n1; h2 = n2;
        s0 += n0; s1 += n1; s2 += n2;
    }
    const float inv = 1.0f / (float)T_SZ;
    hmean[b * H_SZ + 0] = s0 * inv;
    hmean[b * H_SZ + 1] = s1 * inv;
    hmean[b * H_SZ + 2] = s2 * inv;
}

// ---------------------------------------------------------------------------
// Kernel C: out = [st_input | hmean] @ W_fc^T + b_fc  via WMMA f32 16x16x4.
// M=256 (16 tiles) x N=128 (8 tiles), K=35 padded to 36. One wave per tile.
// ---------------------------------------------------------------------------
__device__ __forceinline__ float feat_val(const float* __restrict__ st,
                                          const float* __restrict__ hm,
                                          int m, int k) {
    // feat(m, k) for k in [0, 36): [0,32)=st, [32,35)=hmean, 35=pad(0)
    const int ks = (k < F_ST) ? k : 0;
    int kh = k - F_ST;
    kh = (kh < 0) ? 0 : ((kh > H_SZ - 1) ? H_SZ - 1 : kh);
    const float vs = st[m * F_ST + ks];
    const float vh = hm[m * H_SZ + kh];
    return (k < F_ST) ? vs : ((k < KFC) ? vh : 0.0f);
}

__global__ __launch_bounds__(256) void fc_wmma_kernel(
    const float* __restrict__ st,     // [B, F_ST]
    const float* __restrict__ hmean,  // [B, H]
    const float* __restrict__ W_fc,   // [OUT, KFC]
    const float* __restrict__ b_fc,   // [OUT]
    float* __restrict__ out)          // [B, OUT]
{
    const int lane   = threadIdx.x & 31;
    const int wave   = threadIdx.x >> 5;
    const int tileId = blockIdx.x * 8 + wave;   // 128 tiles = 16 (M) x 8 (N)
    const int mTile  = tileId >> 3;
    const int nTile  = tileId & 7;

    const int half = lane >> 4;
    const int l16  = lane & 15;
    const int m = mTile * 16 + l16;             // batch row (A fragment)
    const int n = nTile * 16 + l16;             // out col   (B fragment)

    v8f c = {};
    #pragma unroll
    for (int k0 = 0; k0 < 36; k0 += 4) {
        const int k = k0 + 2 * half;
        v2f a, b;
        a.x = feat_val(st, hmean, m, k);
        a.y = feat_val(st, hmean, m, k + 1);
        // clamp index, mask value -> no divergence, no OOB on padded K
        const float w0 = W_fc[n * KFC + ((k     < KFC) ? k     : KFC - 1)];
        const float w1 = W_fc[n * KFC + ((k + 1 < KFC) ? k + 1 : KFC - 1)];
        b.x = (k     < KFC) ? w0 : 0.0f;
        b.y = (k + 1 < KFC) ? w1 : 0.0f;
        c = __builtin_amdgcn_wmma_f32_16x16x4_f32(
            false, a, false, b, (short)0, c, false, false);
    }

    const float bias = b_fc[n];
    #pragma unroll
    for (int i = 0; i < 8; ++i) {
        const int row = mTile * 16 + i + 8 * half;
        out[(size_t)row * OUT_SZ + n] = c[i] + bias;
    }
}

// ---------------------------------------------------------------------------
extern "C" void kernel_launch(void* const* d_in, const int* in_sizes, int n_in,
                              void* d_out, int out_size, void* d_ws, size_t ws_size,
                              hipStream_t stream) {
    const float* tr    = (const float*)d_in[0];  // [B, T, F_TR]
    const float* st    = (const float*)d_in[1];  // [B, F_ST]
    const float* W_ih  = (const float*)d_in[2];  // [H, F_TR]
    const float* W_hh  = (const float*)d_in[3];  // [H, H]
    const float* b_ih  = (const float*)d_in[4];  // [H]
    const float* b_hh  = (const float*)d_in[5];  // [H]
    const float* W_fc  = (const float*)d_in[6];  // [OUT, KFC]
    const float* b_fc  = (const float*)d_in[7];  // [OUT]
    float* out = (float*)d_out;                  // [B, OUT]

    float* xproj = (float*)d_ws;                           // T*B*H floats = 3 MB
    float* hmean = xproj + (size_t)T_SZ * B_SZ * H_SZ;     // B*H floats

    // A: 16384 M-tiles, 8 waves/block -> 2048 blocks
    xproj_wmma_kernel<<<(B_SZ * T_SZ / 16) / 8, 256, 0, stream>>>(
        tr, W_ih, b_ih, b_hh, xproj);

    // B: one lane per batch element, 8 waves on one WGP
    rnn_scan_kernel<<<1, 256, 0, stream>>>(xproj, W_hh, hmean);

    // C: 128 tiles, 8 waves/block -> 16 blocks
    fc_wmma_kernel<<<16, 256, 0, stream>>>(st, hmean, W_fc, b_fc, out);
}